// SecGELU_59339268161552
// MI455X (gfx1250) — compile-verified
//
#include <hip/hip_runtime.h>
#include <hip/hip_bf16.h>

// SecGELU for MI455X (gfx1250): memory-bound elementwise LUT op.
// Strategy: stream x with b128 non-temporal loads/stores; stage the 16KB LUT
// into LDS once per block via CDNA5 async global->LDS DMA (ASYNCcnt path),
// gather from LDS (ds_load_b32). Grid-stride so LDS fill amortizes.

#define TBL_N 4096  // 2^(GELU_TABLE_BIT + 2)

typedef float f32x4 __attribute__((ext_vector_type(4)));
typedef int   v4i   __attribute__((__vector_size__(16)));   // matches builtin param type

typedef const __attribute__((address_space(1))) v4i* gv4i_cptr;
typedef __attribute__((address_space(1))) v4i*       gv4i_ptr;
typedef __attribute__((address_space(3))) v4i*       lv4i_ptr;

__global__ __launch_bounds__(256) void secgelu_kernel(
    const float* __restrict__ x,
    const float* __restrict__ table,
    float* __restrict__ out,
    long long n)
{
    __shared__ float tbl[TBL_N];

    // ---- Stage 16KB LUT into LDS -------------------------------------------
    {
        const int t = threadIdx.x;
#if __has_builtin(__builtin_amdgcn_global_load_async_to_lds_b128)
        // CDNA5 async DMA: each lane copies one 16B chunk per issue.
        gv4i_ptr gsrc = (gv4i_ptr)(gv4i_cptr)table;   // drop const, keep AS1
        lv4i_ptr lds  = (lv4i_ptr)tbl;
        // 4096 floats = 1024 x 16B chunks; 256 threads -> 4 chunks each.
#pragma unroll
        for (int i = 0; i < (TBL_N * 4) / (256 * 16); ++i) {
            const int chunk = t + i * 256;
            __builtin_amdgcn_global_load_async_to_lds_b128(
                gsrc + chunk, lds + chunk, /*offset=*/0, /*cpol=*/0);
        }
#if __has_builtin(__builtin_amdgcn_s_wait_asynccnt)
        __builtin_amdgcn_s_wait_asynccnt(0);
#else
        asm volatile("s_wait_asynccnt 0" ::: "memory");
#endif
#else
        // Fallback: copy through VGPRs (16 floats per thread).
#pragma unroll
        for (int i = 0; i < TBL_N / 256; ++i)
            tbl[t + i * 256] = table[t + i * 256];
#endif
        __syncthreads();
    }

    // ---- Streaming main loop: 1 float4 per lane per iteration ---------------
    const long long n4     = n >> 2;
    const long long tid    = (long long)blockIdx.x * blockDim.x + threadIdx.x;
    const long long stride = (long long)gridDim.x * blockDim.x;

    const f32x4* __restrict__ x4 = (const f32x4*)x;
    f32x4* __restrict__       o4 = (f32x4*)out;

    for (long long i = tid; i < n4; i += stride) {
        f32x4 v = __builtin_nontemporal_load(&x4[i]);
        f32x4 r;
#pragma unroll
        for (int k = 0; k < 4; ++k) {
            const float xv = v[k];
            // X = round_rne(x * 2^16); y = floor(X / 64) == Xi >> 6 (arith)
            const int   Xi = __float2int_rn(xv * 65536.0f);
            const int   y  = Xi >> 6;
            const int   a  = (y < 0) ? -y : y;
            const int   c  = (a < TBL_N) ? a : (TBL_N - 1);
            const float dx = (y >= 0) ? xv : 0.0f;
            r[k] = dx - tbl[c];
        }
        __builtin_nontemporal_store(r, &o4[i]);
    }

    // Scalar tail (n is divisible by 4 for this shape; kept for generality).
    for (long long i = (n4 << 2) + tid; i < n; i += stride) {
        const float xv = x[i];
        const int   Xi = __float2int_rn(xv * 65536.0f);
        const int   y  = Xi >> 6;
        const int   a  = (y < 0) ? -y : y;
        const int   c  = (a < TBL_N) ? a : (TBL_N - 1);
        const float dx = (y >= 0) ? xv : 0.0f;
        out[i] = dx - tbl[c];
    }
}

extern "C" void kernel_launch(void* const* d_in, const int* in_sizes, int n_in,
                              void* d_out, int out_size, void* d_ws, size_t ws_size,
                              hipStream_t stream) {
    const float* x     = (const float*)d_in[0];
    const float* table = (const float*)d_in[1];
    float*       out   = (float*)d_out;

    const long long n  = (long long)out_size;   // 4*4096*4096 = 67,108,864
    const long long n4 = (n + 3) >> 2;

    const int block = 256;                      // 8 wave32s
    long long want  = (n4 + block - 1) / block;
    int grid        = (int)((want < 4096) ? (want > 0 ? want : 1) : 4096);

    secgelu_kernel<<<grid, block, 0, stream>>>(x, table, out, n);
}